// Contour_to_distance_map_80839874445331
// MI455X (gfx1250) — compile-verified
//
#include <hip/hip_runtime.h>
#include <math.h>

// Contour-to-distance-map for MI455X (gfx1250, wave32).
// Compute-bound on transcendentals (tanhf/acosf per pixel-vertex pair);
// memory traffic (~0.6 MB) is negligible at 23.3 TB/s. No matmul structure
// (K=2 contraction only), so WMMA does not apply; the relevant CDNA5 feature
// is the async global->LDS copy path (ASYNCcnt) to stage the contour once
// per workgroup, which every wave then reads ~200x per pixel.

#define MAXN 512  // max contour vertices supported (reference uses 200)

__global__ __launch_bounds__(256) void winding_kernel(
    const float* __restrict__ contour, int n, int S,
    float* __restrict__ prod, float* __restrict__ blockmax)
{
    __shared__ __align__(16) float sc[2 * (MAXN + 1)];
    __shared__ float red[256];

    const int tid   = threadIdx.x;
    const int total = 2 * n;          // dwords of contour data
    const int nvec4 = total >> 2;     // float4 chunks (b128 async loads)

    // --- CDNA5 async copy: global -> LDS, tracked by ASYNCcnt -------------
    // Wide b128 body: one float4 per lane.
    for (int i = tid; i < nvec4; i += 256) {
        unsigned lds_off = (unsigned)(size_t)(&sc[4 * i]);
        const float* gp  = contour + 4 * i;
        asm volatile("global_load_async_to_lds_b128 %0, %1, off"
                     :: "v"(lds_off), "v"(gp)
                     : "memory");
    }
    // b32 tail (total not a multiple of 4).
    for (int i = (nvec4 << 2) + tid; i < total; i += 256) {
        unsigned lds_off = (unsigned)(size_t)(&sc[i]);
        const float* gp  = contour + i;
        asm volatile("global_load_async_to_lds_b32 %0, %1, off"
                     :: "v"(lds_off), "v"(gp)
                     : "memory");
    }
    asm volatile("s_wait_asynccnt 0x0" ::: "memory");
    __syncthreads();
    if (tid == 0) {                 // duplicate vertex 0 -> no modulo in hot loop
        sc[total]     = sc[0];
        sc[total + 1] = sc[1];
    }
    __syncthreads();
    // ----------------------------------------------------------------------

    const int  ntot  = S * S;
    const int  p     = blockIdx.x * 256 + tid;
    const bool valid = (p < ntot);

    const float Sf = (float)S;
    const int   pi = p / S;
    const int   pj = p - pi * S;
    const float mx = (float)pi / Sf;   // meshgrid 'ij': first coord = i/S
    const float my = (float)pj / Sf;

    const float eps = 1e-5f;
    const float kk  = 100000.0f;

    // Pipelined edge loop: diff[n+1] is next iteration's diff[n]; its norm
    // (and its eps-clamped value) double as both the vertex distance and the
    // edge's "roll" norm -> one sub-pair + one sqrt + one clamp per edge.
    float dx  = sc[0] - mx;
    float dy  = sc[1] - my;
    float nd  = sqrtf(dx * dx + dy * dy);
    float ndc = fmaxf(nd, eps);

    float w    = 0.0f;
    float mind = nd;

    #pragma unroll 4
    for (int v = 0; v < n; ++v) {
        const float rx  = sc[2 * v + 2] - mx;
        const float ry  = sc[2 * v + 3] - my;
        const float nr  = sqrtf(rx * rx + ry * ry);
        const float nrc = fmaxf(nr, eps);

        // cross[...,1] - cross[...,0] = dy*rx - dx*ry  (matches reference)
        const float cr = dy * rx - dx * ry;
        const float sg = tanhf(kk * cr);

        const float dt = dx * rx + dy * ry;
        float cs = dt / (ndc * nrc);
        cs = fminf(fmaxf(cs, -1.0f + eps), 1.0f - eps);

        w    += sg * acosf(cs);
        mind  = fminf(mind, nd);

        dx = rx; dy = ry; nd = nr; ndc = nrc;
    }

    const float winding = w / 6.283185307179586f;  // / (2*pi), true division
    float pr = winding * mind;

    if (valid) prod[p] = pr;
    else       pr = -INFINITY;

    // Block max reduction (signed max, matching jnp.max)
    red[tid] = pr;
    __syncthreads();
    for (int s2 = 128; s2 > 0; s2 >>= 1) {
        if (tid < s2) red[tid] = fmaxf(red[tid], red[tid + s2]);
        __syncthreads();
    }
    if (tid == 0) blockmax[blockIdx.x] = red[0];
}

__global__ __launch_bounds__(256) void max_reduce_kernel(
    const float* __restrict__ bm, int cnt, float* __restrict__ outmax)
{
    __shared__ float red[256];
    float m = -INFINITY;
    for (int i = threadIdx.x; i < cnt; i += 256) m = fmaxf(m, bm[i]);
    red[threadIdx.x] = m;
    __syncthreads();
    for (int s = 128; s > 0; s >>= 1) {
        if (threadIdx.x < s)
            red[threadIdx.x] = fmaxf(red[threadIdx.x], red[threadIdx.x + s]);
        __syncthreads();
    }
    if (threadIdx.x == 0) *outmax = red[0];
}

__global__ __launch_bounds__(256) void scale_kernel(
    const float* __restrict__ prod, const float* __restrict__ mxp,
    float* __restrict__ out, int total)
{
    const int p = blockIdx.x * 256 + threadIdx.x;
    if (p < total) out[p] = prod[p] / *mxp;   // true division, matches reference
}

extern "C" void kernel_launch(void* const* d_in, const int* in_sizes, int n_in,
                              void* d_out, int out_size, void* d_ws, size_t ws_size,
                              hipStream_t stream)
{
    const float* contour = (const float*)d_in[0];
    const int n = in_sizes[0] / 2;                       // 200 vertices
    const int S = (int)(sqrt((double)out_size) + 0.5);   // 384 (out = 1*1*S*S)
    const int total  = S * S;
    const int blocks = (total + 255) / 256;

    float* ws       = (float*)d_ws;
    float* prod     = ws;                  // total floats
    float* blockmax = ws + total;          // blocks floats
    float* maxv     = ws + total + blocks; // 1 float

    winding_kernel<<<blocks, 256, 0, stream>>>(contour, n, S, prod, blockmax);
    max_reduce_kernel<<<1, 256, 0, stream>>>(blockmax, blocks, maxv);
    scale_kernel<<<blocks, 256, 0, stream>>>(prod, maxv, (float*)d_out, total);
}